// GraphPropagationAttention_11192684773577
// MI455X (gfx1250) — compile-verified
//
#include <hip/hip_runtime.h>
#include <hip/hip_bf16.h>

// ---------------------------------------------------------------------------
// GraphPropagationAttention fused pipeline for gfx1250 (MI455X, wave32, WMMA)
// B=16, N=256, C=512, H=8, E=64, HD=64, SCALE = 1/8
// ---------------------------------------------------------------------------

typedef __attribute__((ext_vector_type(2))) float v2f;
typedef __attribute__((ext_vector_type(8))) float v8f;
typedef int v4i_ __attribute__((vector_size(16)));   // matches builtin proto

#define AS1 __attribute__((address_space(1)))
#define AS3 __attribute__((address_space(3)))

constexpr int cB = 16, cN = 256, cC = 512, cH = 8, cE = 64, cHD = 64;
constexpr float cSCALE = 0.125f;   // 1/sqrt(64)
constexpr float NEG_BIG = -3.0e38f;

__device__ __forceinline__ v8f wmma4(v2f a, v2f b, v8f c) {
  // V_WMMA_F32_16X16X4_F32 : D = A(16x4) * B(4x16) + C(16x16)
  return __builtin_amdgcn_wmma_f32_16x16x4_f32(false, a, false, b, (short)0, c,
                                               false, false);
}

__device__ __forceinline__ void async_b128(const void* gsrc, void* ldst) {
  // GLOBAL_LOAD_ASYNC_TO_LDS_B128 (ASYNCcnt-tracked, bypasses VGPRs)
  __builtin_amdgcn_global_load_async_to_lds_b128(
      (AS1 v4i_*)(AS1 void*)(void*)gsrc, (AS3 v4i_*)(AS3 void*)ldst, 0, 0);
}

// ---------------------------------------------------------------------------
// Kernel 1: QKV projection. Y = X @ W^T,  X:[B*N, C], W:[3C, C]
// One wave per 16x16 output tile. grid(24,16,16) block(128)
// Scatter into q/k/v as [B,H,N,HD].
// ---------------------------------------------------------------------------
__global__ void qkv_kernel(const float* __restrict__ x, const float* __restrict__ w,
                           float* __restrict__ q, float* __restrict__ k,
                           float* __restrict__ v) {
  const int lane = threadIdx.x & 31;
  const int wave = threadIdx.x >> 5;
  const int colTile = blockIdx.x * 4 + wave;   // 0..95 (3C/16)
  const int rowTile = blockIdx.y;              // 0..15 (N/16)
  const int b = blockIdx.z;
  const int idx15 = lane & 15;
  const int khalf = lane >> 4;

  const float* arow = x + ((size_t)(b * cN + rowTile * 16 + idx15)) * cC;
  const int bcol = colTile * 16 + idx15;
  const float* brow = w + (size_t)bcol * cC;   // W[col][k] contiguous in k

  v8f acc = {};
  for (int k0 = 0; k0 < cC; k0 += 4) {
    const int kk = k0 + 2 * khalf;
    v2f a = *(const v2f*)(arow + kk);
    v2f bb = *(const v2f*)(brow + kk);
    acc = wmma4(a, bb, acc);
  }

  const int s = bcol >> 9;        // which of q/k/v
  const int rem = bcol & 511;
  const int h = rem >> 6;
  const int d = rem & 63;
  float* dst = (s == 0) ? q : ((s == 1) ? k : v);
  float* base = dst + ((size_t)(b * cH + h) * cN) * cHD + d;
#pragma unroll
  for (int r = 0; r < 8; ++r) {
    const int row = rowTile * 16 + r + 8 * khalf;
    base[(size_t)row * cHD] = acc[r];
  }
}

// ---------------------------------------------------------------------------
// Kernel 2: fused attention over one (b, 16-row x-strip).
// grid(16,16) block(256) = 8 waves (one wave per head for WMMA phases).
// Phases:
//   A : async-stage q strip; mask strip + small params into LDS
//   B1: attn = SCALE * q k^T  (WMMA, per head)           -> attn_s[8][16][256]
//   B2: attn += reduce(edge)  (edge tiles DOUBLE-BUFFERED via
//       GLOBAL_LOAD_ASYNC_TO_LDS_B128; edge read exactly once from HBM)
//   C : per-row masked max / sumexp
//   D : node partial = softmax(attn) @ v (WMMA, per head) -> node_tmp
//   E0: attn_s <- softmax(attn) + attn   ("g", in place)
//   E : edge_out[e,x,y] = expand(g) (+b), NT stores, coalesced; fused ONLINE
//       pooled softmax over y -> w_pool[b,x,e]
// ---------------------------------------------------------------------------
__global__ void fused_attn_kernel(
    const float* __restrict__ qm, const float* __restrict__ km,
    const float* __restrict__ vm, const float* __restrict__ edge,
    const unsigned char* __restrict__ mask,
    const float* __restrict__ reduce_w, const float* __restrict__ reduce_b,
    const float* __restrict__ expand_w, const float* __restrict__ expand_b,
    float* __restrict__ node_tmp, float* __restrict__ w_pool,
    float* __restrict__ edge_out) {
  extern __shared__ char smem[];
  float* attn_s  = (float*)smem;                      // [8][16][256]  128 KB
  float* q_s     = attn_s + 8 * 16 * 256;             // [8][16][64]    32 KB
  float* edge_s0 = q_s + 8 * 16 * 64;                 // [64][16][16]   64 KB
  float* edge_s1 = edge_s0 + 64 * 16 * 16;            // [64][16][16]   64 KB
  unsigned int* mask_s = (unsigned int*)(edge_s1 + 64 * 16 * 16);  // 4 KB
  float* ms   = (float*)(mask_s + 1024);              // [8][16] row max
  float* is_  = ms + 128;                             // [8][16] 1/sumexp
  float* rw_s = is_ + 128;                            // [8][64]
  float* rb_s = rw_s + 512;                           // [8]
  float* ew_s = rb_s + 8;                             // [64][8]
  float* eb_s = ew_s + 512;                           // [64]

  const int tid  = threadIdx.x;
  const int lane = tid & 31;
  const int wv   = tid >> 5;        // wave id == head id in WMMA phases
  const int idx15 = lane & 15;
  const int khalf = lane >> 4;
  const int b  = blockIdx.y;
  const int x0 = blockIdx.x * 16;

  // ---- Phase A: async-stage q strip; mask strip + small params ----
  for (int j = 0; j < 8; ++j) {
    const int i = tid + 256 * j;            // 2048 float4
    const int h = i >> 8, rem = i & 255, r = rem >> 4, d4 = rem & 15;
    async_b128(qm + ((size_t)(b * cH + h) * cN + x0 + r) * cHD + d4 * 4,
               q_s + (h * 16 + r) * 64 + d4 * 4);
  }
  {
    const unsigned int* mg =
        (const unsigned int*)(mask + (size_t)b * cN * cN + (size_t)x0 * cN);
    for (int j = 0; j < 4; ++j) mask_s[tid + 256 * j] = mg[tid + 256 * j];
  }
  rw_s[tid]       = reduce_w[tid & 511];
  rw_s[tid + 256] = reduce_w[(tid + 256) & 511];
  ew_s[tid]       = expand_w[tid & 511];
  ew_s[tid + 256] = expand_w[(tid + 256) & 511];
  if (tid < 64) eb_s[tid] = expand_b[tid];
  if (tid < 8)  rb_s[tid] = reduce_b[tid];
  __builtin_amdgcn_s_wait_asynccnt(0);
  __syncthreads();

  // ---- Phase B1: attn = SCALE * q @ k^T  per head (wave wv) ----
  {
    const int h = wv;
    const float* qrow = q_s + (h * 16 + idx15) * 64;
    for (int y0 = 0; y0 < cN; y0 += 16) {
      const float* kcol = km + ((size_t)(b * cH + h) * cN + y0 + idx15) * cHD;
      v8f acc = {};
      for (int k0 = 0; k0 < cHD; k0 += 4) {
        const int kk = k0 + 2 * khalf;
        v2f a  = *(const v2f*)(qrow + kk);
        v2f bb = *(const v2f*)(kcol + kk);
        acc = wmma4(a, bb, acc);
      }
#pragma unroll
      for (int r = 0; r < 8; ++r) {
        const int row = r + 8 * khalf;
        attn_s[(h * 16 + row) * 256 + y0 + idx15] = acc[r] * cSCALE;
      }
    }
  }
  __syncthreads();

  // ---- Phase B2: attn += per-head bias from edge_embeds ----
  // Edge tiles stream via async-to-LDS with double buffering: the next 64 KB
  // tile is in flight while the current one feeds the reduce_w contraction.
  {
    // prologue: issue tile y0=0 into edge_s0 (16 b128 per thread)
    for (int j = 0; j < 16; ++j) {
      const int i = tid + 256 * j;
      const int e = i >> 6, rem = i & 63, rr = rem >> 2, cc = (rem & 3) * 4;
      async_b128(edge + ((size_t)(b * cE + e) * cN + x0 + rr) * cN + 0 + cc,
                 edge_s0 + (e * 16 + rr) * 16 + cc);
    }
    for (int y0 = 0; y0 < cN; y0 += 16) {
      float* cur = (y0 & 16) ? edge_s1 : edge_s0;
      float* nxt = (y0 & 16) ? edge_s0 : edge_s1;
      if (y0 + 16 < cN) {
        for (int j = 0; j < 16; ++j) {
          const int i = tid + 256 * j;
          const int e = i >> 6, rem = i & 63, rr = rem >> 2, cc = (rem & 3) * 4;
          async_b128(
              edge + ((size_t)(b * cE + e) * cN + x0 + rr) * cN + y0 + 16 + cc,
              nxt + (e * 16 + rr) * 16 + cc);
        }
        __builtin_amdgcn_s_wait_asynccnt(16);  // cur's 16 loads retired
      } else {
        __builtin_amdgcn_s_wait_asynccnt(0);
      }
      __syncthreads();                          // cur visible to all waves
      for (int j = 0; j < 8; ++j) {             // 2048 bias outputs
        const int i = tid + 256 * j;
        const int hh = i >> 8, pos = i & 255, rr = pos >> 4, cc = pos & 15;
        float sum = rb_s[hh];
        for (int e = 0; e < cE; ++e)
          sum += rw_s[hh * 64 + e] * cur[(e * 16 + rr) * 16 + cc];
        attn_s[(hh * 16 + rr) * 256 + y0 + cc] += sum;
      }
      __syncthreads();                          // done reading cur
    }
  }

  // ---- Phase C: masked row max / 1/sumexp (wave wv handles head wv) ----
  {
    const int h = wv;
    const int row = idx15;
    const unsigned char* mrow = (const unsigned char*)mask_s + row * 256;
    const float* arow = attn_s + (h * 16 + row) * 256;
    float mx = NEG_BIG;
    for (int y = khalf * 128; y < khalf * 128 + 128; ++y) {
      const float val = mrow[y] ? NEG_BIG : arow[y];
      mx = fmaxf(mx, val);
    }
    mx = fmaxf(mx, __shfl_xor(mx, 16, 32));
    float se = 0.f;
    for (int y = khalf * 128; y < khalf * 128 + 128; ++y)
      if (!mrow[y]) se += __expf(arow[y] - mx);
    se += __shfl_xor(se, 16, 32);
    if (lane < 16) { ms[h * 16 + row] = mx; is_[h * 16 + row] = 1.0f / se; }
  }
  __syncthreads();

  // ---- Phase D: node partial = softmax(attn) @ v  (WMMA, per head) ----
  {
    const int h = wv;
    const float mrow = ms[h * 16 + idx15];
    const float irow = is_[h * 16 + idx15];
    const float* arow = attn_s + (h * 16 + idx15) * 256;
    const unsigned char* mbytes = (const unsigned char*)mask_s + idx15 * 256;
    const float* vbase = vm + ((size_t)(b * cH + h) * cN) * cHD;
    v8f acc0 = {}, acc1 = {}, acc2 = {}, acc3 = {};
    for (int k0 = 0; k0 < cN; k0 += 4) {
      const int kk = k0 + 2 * khalf;
      v2f a;
      a.x = mbytes[kk]     ? 0.f : __expf(arow[kk]     - mrow) * irow;
      a.y = mbytes[kk + 1] ? 0.f : __expf(arow[kk + 1] - mrow) * irow;
      const float* v0 = vbase + (size_t)kk * cHD;
      v2f b0 = { v0[idx15],      v0[cHD + idx15] };
      v2f b1 = { v0[16 + idx15], v0[cHD + 16 + idx15] };
      v2f b2 = { v0[32 + idx15], v0[cHD + 32 + idx15] };
      v2f b3 = { v0[48 + idx15], v0[cHD + 48 + idx15] };
      acc0 = wmma4(a, b0, acc0);
      acc1 = wmma4(a, b1, acc1);
      acc2 = wmma4(a, b2, acc2);
      acc3 = wmma4(a, b3, acc3);
    }
    float* nbase = node_tmp + ((size_t)(b * cN + x0)) * cC + h * cHD;
#pragma unroll
    for (int r = 0; r < 8; ++r) {
      const int row = r + 8 * khalf;
      nbase[(size_t)row * cC +  0 + idx15] = acc0[r];
      nbase[(size_t)row * cC + 16 + idx15] = acc1[r];
      nbase[(size_t)row * cC + 32 + idx15] = acc2[r];
      nbase[(size_t)row * cC + 48 + idx15] = acc3[r];
    }
  }
  __syncthreads();

  // ---- Phase E0: attn_s <- softmax(attn) + attn  ("g", in place) ----
  for (int i = tid; i < 8 * 16 * 256; i += 256) {
    const int hh = i >> 12, rem = i & 4095, r = rem >> 8, y = rem & 255;
    const float val = attn_s[i];
    const float sm = ((const unsigned char*)mask_s)[r * 256 + y]
                         ? 0.f
                         : __expf(val - ms[hh * 16 + r]) * is_[hh * 16 + r];
    attn_s[i] = sm + val;
  }
  __syncthreads();

  // ---- Phase E: expand to edge_out + fused online pooled softmax ----
  // wave wv owns x = 2*wv, 2*wv+1; 32 lanes split y (coalesced 128B NT
  // stores to edge_out); per-lane g[8h][8y] cached in registers.
  for (int xi = 0; xi < 2; ++xi) {
    const int x = wv * 2 + xi;
    float g[8][8];
#pragma unroll
    for (int hh = 0; hh < 8; ++hh)
#pragma unroll
      for (int j = 0; j < 8; ++j)
        g[hh][j] = attn_s[(hh * 16 + x) * 256 + lane + 32 * j];
    unsigned char mk[8];
#pragma unroll
    for (int j = 0; j < 8; ++j)
      mk[j] = ((const unsigned char*)mask_s)[x * 256 + lane + 32 * j];

    for (int e = 0; e < cE; ++e) {
      float ew[8];
#pragma unroll
      for (int hh = 0; hh < 8; ++hh) ew[hh] = ew_s[e * 8 + hh];
      const float eb = eb_s[e];
      float mm = NEG_BIG, se = 0.f, sv = 0.f;
      float* eo = edge_out + ((size_t)(b * cE + e) * cN + x0 + x) * cN;
#pragma unroll
      for (int j = 0; j < 8; ++j) {
        float t = eb;
#pragma unroll
        for (int hh = 0; hh < 8; ++hh) t += ew[hh] * g[hh][j];
        __builtin_nontemporal_store(t, eo + lane + 32 * j);  // streamed, NT
        if (!mk[j]) {
          if (t > mm) { const float sc = __expf(mm - t); se *= sc; sv *= sc; mm = t; }
          const float p = __expf(t - mm);
          se += p; sv += p * t;
        }
      }
      // cross-lane online-softmax merge
      for (int off = 16; off > 0; off >>= 1) {
        const float mm2 = __shfl_xor(mm, off, 32);
        const float se2 = __shfl_xor(se, off, 32);
        const float sv2 = __shfl_xor(sv, off, 32);
        const float mmax = fmaxf(mm, mm2);
        const float s1 = __expf(mm - mmax), s2 = __expf(mm2 - mmax);
        se = se * s1 + se2 * s2;
        sv = sv * s1 + sv2 * s2;
        mm = mmax;
      }
      if (lane == 0)
        w_pool[((size_t)(b * cN + x0 + x)) * cE + e] = sv / se;
    }
  }
}

// ---------------------------------------------------------------------------
// Kernel 3: t = node_tmp + w_pool @ fc_w^T + fc_b   (rows=4096, cols=512, K=64)
// grid(8,256) block(128)
// ---------------------------------------------------------------------------
__global__ void fc_kernel(const float* __restrict__ node_tmp,
                          const float* __restrict__ w_pool,
                          const float* __restrict__ fc_w,
                          const float* __restrict__ fc_b,
                          float* __restrict__ out_tmp) {
  const int lane = threadIdx.x & 31;
  const int wave = threadIdx.x >> 5;
  const int colTile = blockIdx.x * 4 + wave;   // 0..31
  const int rowTile = blockIdx.y;              // 0..255
  const int idx15 = lane & 15;
  const int khalf = lane >> 4;

  const float* arow = w_pool + (size_t)(rowTile * 16 + idx15) * cE;
  const float* brow = fc_w + (size_t)(colTile * 16 + idx15) * cE;
  v8f acc = {};
  for (int k0 = 0; k0 < cE; k0 += 4) {
    const int kk = k0 + 2 * khalf;
    acc = wmma4(*(const v2f*)(arow + kk), *(const v2f*)(brow + kk), acc);
  }
  const int col = colTile * 16 + idx15;
  const float bias = fc_b[col];
#pragma unroll
  for (int r = 0; r < 8; ++r) {
    const size_t row = rowTile * 16 + r + 8 * khalf;
    out_tmp[row * cC + col] = acc[r] + bias + node_tmp[row * cC + col];
  }
}

// ---------------------------------------------------------------------------
// Kernel 4: node_out = t @ proj_w^T + proj_b  (rows=4096, cols=512, K=512)
// grid(8,256) block(128)
// ---------------------------------------------------------------------------
__global__ void proj_kernel(const float* __restrict__ x,
                            const float* __restrict__ w,
                            const float* __restrict__ bias,
                            float* __restrict__ out) {
  const int lane = threadIdx.x & 31;
  const int wave = threadIdx.x >> 5;
  const int colTile = blockIdx.x * 4 + wave;   // 0..31
  const int rowTile = blockIdx.y;              // 0..255
  const int idx15 = lane & 15;
  const int khalf = lane >> 4;

  const float* arow = x + (size_t)(rowTile * 16 + idx15) * cC;
  const float* brow = w + (size_t)(colTile * 16 + idx15) * cC;
  v8f acc = {};
  for (int k0 = 0; k0 < cC; k0 += 4) {
    const int kk = k0 + 2 * khalf;
    acc = wmma4(*(const v2f*)(arow + kk), *(const v2f*)(brow + kk), acc);
  }
  const int col = colTile * 16 + idx15;
  const float bb = bias[col];
#pragma unroll
  for (int r = 0; r < 8; ++r) {
    const size_t row = rowTile * 16 + r + 8 * khalf;
    out[row * cC + col] = acc[r] + bb;
  }
}

// ---------------------------------------------------------------------------
extern "C" void kernel_launch(void* const* d_in, const int* in_sizes, int n_in,
                              void* d_out, int out_size, void* d_ws,
                              size_t ws_size, hipStream_t stream) {
  (void)in_sizes; (void)n_in; (void)out_size; (void)ws_size;

  const float* node_embeds        = (const float*)d_in[0];
  const float* edge_embeds        = (const float*)d_in[1];
  const unsigned char* padding_mask = (const unsigned char*)d_in[2];
  const float* qkv_w   = (const float*)d_in[3];
  const float* proj_w  = (const float*)d_in[4];
  const float* proj_b  = (const float*)d_in[5];
  const float* reduce_w = (const float*)d_in[6];
  const float* reduce_b = (const float*)d_in[7];
  const float* expand_w = (const float*)d_in[8];
  const float* expand_b = (const float*)d_in[9];
  const float* fc_w = (const float*)d_in[10];
  const float* fc_b = (const float*)d_in[11];

  float* node_out = (float*)d_out;                        // [B,N,C]
  float* edge_out = node_out + (size_t)cB * cN * cC;      // [B,E,N,N]

  constexpr size_t QSZ = (size_t)cB * cH * cN * cHD;      // 2,097,152 floats
  float* q        = (float*)d_ws;
  float* k        = q + QSZ;
  float* v        = k + QSZ;
  float* node_tmp = v + QSZ;                              // [B,N,C]
  float* w_pool   = node_tmp + QSZ;                       // [B,N,E]
  float* node_tmp2 = q;   // q dead after fused kernel; reuse for fc output

  qkv_kernel<<<dim3(24, 16, 16), 128, 0, stream>>>(node_embeds, qkv_w, q, k, v);

  constexpr size_t SMEM_BYTES =
      (size_t)4 * (8 * 16 * 256 + 8 * 16 * 64 + 2 * 64 * 16 * 16 + 1024 +
                   128 + 128 + 512 + 8 + 512 + 64);       // 304,416 B
  fused_attn_kernel<<<dim3(16, 16), 256, SMEM_BYTES, stream>>>(
      q, k, v, edge_embeds, padding_mask, reduce_w, reduce_b, expand_w,
      expand_b, node_tmp, w_pool, edge_out);

  fc_kernel<<<dim3(8, 256), 128, 0, stream>>>(node_tmp, w_pool, fc_w, fc_b,
                                              node_tmp2);
  proj_kernel<<<dim3(8, 256), 128, 0, stream>>>(node_tmp2, proj_w, proj_b,
                                                node_out);
}